// OptimizedGCN_84997402787955
// MI455X (gfx1250) — compile-verified
//
#include <hip/hip_runtime.h>
#include <hip/hip_bf16.h>

// ---------------------------------------------------------------------------
// GCN pipeline for MI455X (gfx1250, wave32, WMMA + async-to-LDS).
//   conv1:  h = x @ W1                      (K=4  -> 1 WMMA per 16x16 tile)
//   agg1 :  out[i] = sum_e norm*h[src] + dinv[i]^2*h[i] + b1   (scatter-add)
//   LN+ReLU
//   conv2:  h @ W2, agg2, LN+ReLU
//   mlp  :  relu(h@Wf1+bf1) @ Wf2 + bf2     (K=64 -> 16 WMMAs per tile)
// Dense GEMMs use V_WMMA_F32_16X16X4_F32 (fp32 in/out); A/W tiles staged to
// LDS via GLOBAL_LOAD_ASYNC_TO_LDS_B128 (ASYNCcnt) when available.
// ---------------------------------------------------------------------------

typedef __attribute__((ext_vector_type(2))) float v2f;
typedef __attribute__((ext_vector_type(8))) float v8f;
typedef __attribute__((ext_vector_type(4))) int   v4i;

#define HID 64

#if defined(__has_builtin)
#if __has_builtin(__builtin_amdgcn_global_load_async_to_lds_b128)
#define GCN_ASYNC_LDS 1
#endif
#endif

#ifdef GCN_ASYNC_LDS
typedef __attribute__((address_space(1))) v4i* g_v4i_p;   // global int4*
typedef __attribute__((address_space(3))) v4i* l_v4i_p;   // LDS int4*

__device__ __forceinline__ void gcn_async_cp16(const float* gsrc, float* lds) {
    __builtin_amdgcn_global_load_async_to_lds_b128((g_v4i_p)gsrc,
                                                   (l_v4i_p)lds, 0, 0);
}
__device__ __forceinline__ void gcn_async_wait() {
#if __has_builtin(__builtin_amdgcn_s_wait_asynccnt)
    __builtin_amdgcn_s_wait_asynccnt(0);
#else
    asm volatile("s_wait_asynccnt 0x0" ::: "memory");
#endif
}
#endif

// ---------------------------- degree / norm --------------------------------

__global__ void gcn_deg_init(float* __restrict__ deg, int n) {
    int i = blockIdx.x * blockDim.x + threadIdx.x;
    if (i < n) deg[i] = 1.0f;   // self-loop contributes 1 to degree
}

__global__ void gcn_deg_count(const int* __restrict__ ei, float* __restrict__ deg,
                              int n_edges) {
    int e = blockIdx.x * blockDim.x + threadIdx.x;
    if (e < n_edges) {
        int dst = ei[n_edges + e];
        atomicAdd(&deg[dst], 1.0f);
    }
}

__global__ void gcn_deg_rsqrt(float* __restrict__ deg, int n) {
    int i = blockIdx.x * blockDim.x + threadIdx.x;
    if (i < n) deg[i] = rsqrtf(deg[i]);   // deg >= 1 always (self loop)
}

// ------------------------- GEMM, K = 4 (x @ W1) ----------------------------
// A: [n_rows x 4], W: [4 x 64], C: [n_rows x 64].
// Block = 256 thr = 8 waves; block covers 32 rows x 64 cols = 8 tiles of 16x16.

__global__ __launch_bounds__(256) void gcn_gemm_k4(
    const float* __restrict__ A, const float* __restrict__ W,
    float* __restrict__ C, int n_rows) {
    __shared__ float Wlds[4 * 64];
    __shared__ float Alds[32 * 4];

    const int tid = threadIdx.x;
    const int rowBase = blockIdx.x * 32;

    Wlds[tid] = W[tid];                        // 256 floats, 1 per thread
    if (tid < 128) {                           // 32 rows x 4 cols
        int r = rowBase + (tid >> 2);
        if (r >= n_rows) r = n_rows - 1;       // clamp: masked rows only
        Alds[tid] = A[r * 4 + (tid & 3)];
    }
    __syncthreads();

    const int wave = tid >> 5;                 // 0..7
    const int lane = tid & 31;
    const int rowTile = (wave >> 2) << 4;      // 0 or 16
    const int colTile = (wave & 3) << 4;       // 0,16,32,48
    const int m = lane & 15;
    const int khalf = (lane >> 4) << 1;        // lanes 0-15 -> K0/K1, 16-31 -> K2/K3

    v2f a, b;
    a.x = Alds[(rowTile + m) * 4 + khalf];
    a.y = Alds[(rowTile + m) * 4 + khalf + 1];
    b.x = Wlds[(khalf)     * 64 + colTile + m];
    b.y = Wlds[(khalf + 1) * 64 + colTile + m];

    v8f acc = {};
    acc = __builtin_amdgcn_wmma_f32_16x16x4_f32(false, a, false, b,
                                                (short)0, acc, false, false);

    const int nOut  = colTile + m;
    const int mRow  = rowBase + rowTile + ((lane >> 4) << 3);
    if (rowBase + 32 <= n_rows) {              // uniform fast path: no guards
#pragma unroll
        for (int r = 0; r < 8; ++r) C[(mRow + r) * HID + nOut] = acc[r];
    } else {
#pragma unroll
        for (int r = 0; r < 8; ++r)
            if (mRow + r < n_rows) C[(mRow + r) * HID + nOut] = acc[r];
    }
}

// ------------------------- GEMM, K = 64 ------------------------------------
// A: [n_rows x 64], W: [64 x 64], optional bias[64], optional ReLU.
// 16 chained V_WMMA_F32_16X16X4_F32 per wave tile; LDS staged with async b128.

__global__ __launch_bounds__(256) void gcn_gemm_k64(
    const float* __restrict__ A, const float* __restrict__ W,
    const float* __restrict__ bias, float* __restrict__ C,
    int n_rows, int relu) {
    __shared__ float Wlds[64 * 64];
    __shared__ float Alds[32 * 64];

    const int tid = threadIdx.x;
    const int rowBase = blockIdx.x * 32;

#ifdef GCN_ASYNC_LDS
    // W: 64*64 floats = 1024 x 16B chunks; 4 per thread.
#pragma unroll
    for (int j = 0; j < 4; ++j) {
        int c = tid + j * 256;
        gcn_async_cp16(W + c * 4, Wlds + c * 4);
    }
    // A: 32*64 floats = 512 x 16B chunks; 2 per thread; clamp row for tails.
#pragma unroll
    for (int j = 0; j < 2; ++j) {
        int c = tid + j * 256;
        int r = rowBase + (c >> 4);
        if (r >= n_rows) r = n_rows - 1;       // clamp: masked rows only
        gcn_async_cp16(A + (size_t)r * HID + ((c & 15) << 2), Alds + c * 4);
    }
    gcn_async_wait();
    __syncthreads();
#else
    for (int i = tid; i < 64 * 64; i += 256) Wlds[i] = W[i];
    for (int i = tid; i < 32 * 64; i += 256) {
        int r = rowBase + (i >> 6);
        if (r >= n_rows) r = n_rows - 1;
        Alds[i] = A[r * HID + (i & 63)];
    }
    __syncthreads();
#endif

    const int wave = tid >> 5;
    const int lane = tid & 31;
    const int rowTile = (wave >> 2) << 4;
    const int colTile = (wave & 3) << 4;
    const int m = lane & 15;
    const int khalf = (lane >> 4) << 1;

    v8f acc = {};
#pragma unroll
    for (int k0 = 0; k0 < 64; k0 += 4) {
        v2f a, b;
        a.x = Alds[(rowTile + m) * 64 + k0 + khalf];
        a.y = Alds[(rowTile + m) * 64 + k0 + khalf + 1];
        b.x = Wlds[(k0 + khalf)     * 64 + colTile + m];
        b.y = Wlds[(k0 + khalf + 1) * 64 + colTile + m];
        acc = __builtin_amdgcn_wmma_f32_16x16x4_f32(false, a, false, b,
                                                    (short)0, acc, false, false);
    }

    const int nOut  = colTile + m;
    const int mRow  = rowBase + rowTile + ((lane >> 4) << 3);
    const float bv  = bias ? bias[nOut] : 0.0f;
    if (rowBase + 32 <= n_rows) {              // uniform fast path: no guards
#pragma unroll
        for (int r = 0; r < 8; ++r) {
            float v = acc[r] + bv;
            if (relu) v = fmaxf(v, 0.0f);
            C[(mRow + r) * HID + nOut] = v;
        }
    } else {
#pragma unroll
        for (int r = 0; r < 8; ++r) {
            if (mRow + r < n_rows) {
                float v = acc[r] + bv;
                if (relu) v = fmaxf(v, 0.0f);
                C[(mRow + r) * HID + nOut] = v;
            }
        }
    }
}

// -------------------- self-loop + bias init of aggregation -----------------
// out[i, f] = dinv[i]^2 * h[i, f] + b[f]

__global__ void gcn_selfloop_bias(const float* __restrict__ h,
                                  const float* __restrict__ dinv,
                                  const float* __restrict__ b,
                                  float* __restrict__ out, int n_nodes) {
    int i = blockIdx.x * blockDim.x + threadIdx.x;
    if (i < n_nodes * HID) {
        int node = i >> 6;
        float d = dinv[node];
        out[i] = h[i] * d * d + b[i & 63];
    }
}

// ----------------------- edge scatter-add aggregation ----------------------
// 16 lanes per edge, float4 per lane: out[dst] += dinv[src]*dinv[dst]*h[src]

__global__ void gcn_aggregate(const float* __restrict__ h,
                              const float* __restrict__ dinv,
                              const int* __restrict__ ei,
                              float* __restrict__ out, int n_edges) {
    long long gid = (long long)blockIdx.x * blockDim.x + threadIdx.x;
    long long e = gid >> 4;
    if (e >= n_edges) return;
    int f0 = (int)(gid & 15) << 2;
    int src = ei[e];
    int dst = ei[n_edges + e];
    float norm = dinv[src] * dinv[dst];
    const float4 hv = *(const float4*)(h + (long long)src * HID + f0);
    float* o = out + (long long)dst * HID + f0;
    atomicAdd(o + 0, hv.x * norm);
    atomicAdd(o + 1, hv.y * norm);
    atomicAdd(o + 2, hv.z * norm);
    atomicAdd(o + 3, hv.w * norm);
}

// -------------------------- LayerNorm + ReLU -------------------------------
// wave32-native: 32 lanes per node, 2 features per lane, shfl_xor reductions.

__global__ __launch_bounds__(256) void gcn_ln_relu(
    const float* __restrict__ in, const float* __restrict__ g,
    const float* __restrict__ beta, float* __restrict__ out, int n_nodes) {
    const int lane = threadIdx.x & 31;
    const int node = (blockIdx.x * blockDim.x + threadIdx.x) >> 5;
    if (node >= n_nodes) return;

    const float x0 = in[node * HID + lane];
    const float x1 = in[node * HID + lane + 32];
    float s = x0 + x1;
    float q = x0 * x0 + x1 * x1;
#pragma unroll
    for (int off = 16; off >= 1; off >>= 1) {
        s += __shfl_xor(s, off, 32);
        q += __shfl_xor(q, off, 32);
    }
    const float mu   = s * (1.0f / 64.0f);
    const float var  = q * (1.0f / 64.0f) - mu * mu;
    const float rstd = rsqrtf(var + 1e-5f);

    float y0 = (x0 - mu) * rstd * g[lane]      + beta[lane];
    float y1 = (x1 - mu) * rstd * g[lane + 32] + beta[lane + 32];
    out[node * HID + lane]      = fmaxf(y0, 0.0f);
    out[node * HID + lane + 32] = fmaxf(y1, 0.0f);
}

// ---------------------------------------------------------------------------

extern "C" void kernel_launch(void* const* d_in, const int* in_sizes, int n_in,
                              void* d_out, int out_size, void* d_ws, size_t ws_size,
                              hipStream_t stream) {
    const float* x    = (const float*)d_in[0];
    const int*   ei   = (const int*)  d_in[1];
    const float* W1   = (const float*)d_in[2];
    const float* b1   = (const float*)d_in[3];
    const float* g1   = (const float*)d_in[4];
    const float* be1  = (const float*)d_in[5];
    const float* W2   = (const float*)d_in[6];
    const float* b2   = (const float*)d_in[7];
    const float* g2   = (const float*)d_in[8];
    const float* be2  = (const float*)d_in[9];
    const float* Wf1  = (const float*)d_in[10];
    const float* bf1  = (const float*)d_in[11];
    const float* Wf2  = (const float*)d_in[12];
    const float* bf2  = (const float*)d_in[13];
    float* out = (float*)d_out;

    const int n_nodes = in_sizes[0] / 4;       // x is [N, 4]
    const int n_edges = in_sizes[1] / 2;       // edge_index is [2, E]
    const long long nf = (long long)n_nodes * HID;

    // workspace carve-out: dinv | bufA | bufB
    float* dinv = (float*)d_ws;
    float* bufA = dinv + (((size_t)n_nodes + 255) & ~(size_t)255);
    float* bufB = bufA + (((size_t)nf + 255) & ~(size_t)255);

    const int T = 256;
    const dim3 blkN((n_nodes + T - 1) / T);
    const dim3 blkNF((unsigned)((nf + T - 1) / T));
    const dim3 blkE((n_edges + T - 1) / T);
    const dim3 blkE16((unsigned)(((long long)n_edges * 16 + T - 1) / T));
    const dim3 blkTile((n_nodes + 31) / 32);
    const dim3 blkLN((n_nodes * 32 + T - 1) / T);

    // --- degrees -> dinv ---
    gcn_deg_init <<<blkN,  T, 0, stream>>>(dinv, n_nodes);
    gcn_deg_count<<<blkE,  T, 0, stream>>>(ei, dinv, n_edges);
    gcn_deg_rsqrt<<<blkN,  T, 0, stream>>>(dinv, n_nodes);

    // --- conv1 ---
    gcn_gemm_k4      <<<blkTile, T, 0, stream>>>(x, W1, bufA, n_nodes);
    gcn_selfloop_bias<<<blkNF,   T, 0, stream>>>(bufA, dinv, b1, bufB, n_nodes);
    gcn_aggregate    <<<blkE16,  T, 0, stream>>>(bufA, dinv, ei, bufB, n_edges);
    gcn_ln_relu      <<<blkLN,   T, 0, stream>>>(bufB, g1, be1, bufA, n_nodes);

    // --- conv2 ---
    gcn_gemm_k64     <<<blkTile, T, 0, stream>>>(bufA, W2, nullptr, bufB, n_nodes, 0);
    gcn_selfloop_bias<<<blkNF,   T, 0, stream>>>(bufB, dinv, b2, bufA, n_nodes);
    gcn_aggregate    <<<blkE16,  T, 0, stream>>>(bufB, dinv, ei, bufA, n_edges);
    gcn_ln_relu      <<<blkLN,   T, 0, stream>>>(bufA, g2, be2, bufB, n_nodes);

    // --- MLP head ---
    gcn_gemm_k64<<<blkTile, T, 0, stream>>>(bufB, Wf1, bf1, bufA, n_nodes, 1);
    gcn_gemm_k64<<<blkTile, T, 0, stream>>>(bufA, Wf2, bf2, out,  n_nodes, 0);
}